// FRFN_38482906972647
// MI455X (gfx1250) — compile-verified
//
#include <hip/hip_runtime.h>
#include <math.h>

// ---------------- problem constants (from reference) ----------------
#define BATCH  64
#define DIM_C  256          // input channels
#define HID_C  680          // int(256*2.66)
#define CH_C   1360         // 2*HID
#define PIX    196          // 14*14
#define NTOT   (BATCH*PIX)  // 12544  (flattened batch*pixels = GEMM N)
#define INTER  64
#define OUTF   (CH_C*9)     // 12240  (final conv out channels)
#define KIM    (INTER*9)    // 576    (im2col K)

typedef __attribute__((ext_vector_type(2))) float v2f;
typedef __attribute__((ext_vector_type(8))) float v8f;

// ---------------- pack x (b,c,p) -> Xp (c, b*196+p) ----------------
__global__ __launch_bounds__(256) void k_pack_x(const float* __restrict__ x,
                                                float* __restrict__ Xp) {
  int gid = blockIdx.x * blockDim.x + threadIdx.x;
  if (gid >= BATCH * DIM_C * PIX) return;
  int b = gid / (DIM_C * PIX);
  int r = gid % (DIM_C * PIX);
  int c = r / PIX;
  int p = r % PIX;
  Xp[(size_t)c * NTOT + b * PIX + p] = x[gid];
}

// ---------------- unpack Out2 (o, b*196+p) -> out (b,o,p) ----------------
__global__ __launch_bounds__(256) void k_unpack(const float* __restrict__ O2,
                                                float* __restrict__ out) {
  int gid = blockIdx.x * blockDim.x + threadIdx.x;
  if (gid >= BATCH * DIM_C * PIX) return;
  int b = gid / (DIM_C * PIX);
  int r = gid % (DIM_C * PIX);
  int o = r / PIX;
  int p = r % PIX;
  out[gid] = O2[(size_t)o * NTOT + b * PIX + p];
}

// ---- conv3x3(pad1) + LayerNorm over (C,H,W) + ReLU; single workgroup ----
// in: (CI,196), w: (64,CI,3,3), g/bb: (64,196), tmp/out: (64,196)
template <int CI>
__global__ __launch_bounds__(256) void k_conv_ln(const float* __restrict__ in,
                                                 const float* __restrict__ w,
                                                 const float* __restrict__ g,
                                                 const float* __restrict__ bb,
                                                 float* __restrict__ tmp,
                                                 float* __restrict__ out) {
  __shared__ float s1[256];
  __shared__ float s2[256];
  float lsum = 0.f, lsq = 0.f;
  for (int idx = threadIdx.x; idx < INTER * PIX; idx += 256) {
    int o = idx / PIX, p = idx % PIX;
    int h = p / 14, wc = p % 14;
    float acc = 0.f;
    for (int ci = 0; ci < CI; ++ci) {
#pragma unroll
      for (int i = 0; i < 3; ++i) {
        int hh = h + i - 1;
        if (hh < 0 || hh >= 14) continue;
#pragma unroll
        for (int j = 0; j < 3; ++j) {
          int ww = wc + j - 1;
          if (ww < 0 || ww >= 14) continue;
          acc += w[((o * CI + ci) * 3 + i) * 3 + j] * in[ci * PIX + hh * 14 + ww];
        }
      }
    }
    tmp[idx] = acc;
    lsum += acc;
    lsq += acc * acc;
  }
  s1[threadIdx.x] = lsum;
  s2[threadIdx.x] = lsq;
  __syncthreads();
  for (int s = 128; s > 0; s >>= 1) {
    if ((int)threadIdx.x < s) {
      s1[threadIdx.x] += s1[threadIdx.x + s];
      s2[threadIdx.x] += s2[threadIdx.x + s];
    }
    __syncthreads();
  }
  const float inv_n = 1.0f / (float)(INTER * PIX);
  float mean = s1[0] * inv_n;
  float var  = s2[0] * inv_n - mean * mean;
  float rstd = rsqrtf(var + 1e-5f);
  for (int idx = threadIdx.x; idx < INTER * PIX; idx += 256) {
    float v = (tmp[idx] - mean) * rstd * g[idx] + bb[idx];
    out[idx] = v > 0.f ? v : 0.f;
  }
}

// ---------------- im2col for final 3x3 conv: t2(64,196) -> Bf(576,196) ----
__global__ __launch_bounds__(256) void k_im2col(const float* __restrict__ t2,
                                                float* __restrict__ Bf) {
  int gid = blockIdx.x * blockDim.x + threadIdx.x;
  if (gid >= KIM * PIX) return;
  int row = gid / PIX, p = gid % PIX;
  int c = row / 9, k = row % 9;
  int i = k / 3, j = k % 3;
  int h = p / 14, wc = p % 14;
  int hh = h + i - 1, ww = wc + j - 1;
  float v = 0.f;
  if (hh >= 0 && hh < 14 && ww >= 0 && ww < 14) v = t2[c * PIX + hh * 14 + ww];
  Bf[gid] = v;
}

// ---------------- f32 WMMA GEMM: C(MxN) = A(MxK) * B(KxN), row-major ------
// M, N, Kd are compile-time so the backend can fold strides into immediate
// offsets and strength-reduce induction variables. K % 4 == 0 required.
// Edges via index clamping (loads) + guarded stores: branch-free inner loop.
// Block = 256 threads = 8 waves arranged 4(M) x 2(N).
// Wave tile = 64(M) x 16(N): 4 accumulators, B fragment reused across the
// 4 M-subtiles -> per K-step: 4x b64 A-loads + 2x b32 B-loads + 4x WMMA.
template <int M, int N, int Kd>
__global__ __launch_bounds__(256) void k_gemm_wmma(const float* __restrict__ A,
                                                   const float* __restrict__ B,
                                                   float* __restrict__ C) {
  const int lane  = threadIdx.x & 31;
  const int wave  = threadIdx.x >> 5;
  const int wm    = wave & 3;   // wave position along M (4)
  const int wn    = wave >> 2;  // wave position along N (2)
  const int m0    = blockIdx.y * 256 + wm * 64;
  if (m0 >= M) return;          // wave-uniform exit
  const int idx16 = lane & 15;        // M-row (A frag) / N-col (B frag)
  const int khalf = (lane >> 4) * 2;  // K sub-offset: 0 or 2

  const int col  = blockIdx.x * 32 + wn * 16 + idx16;
  const int colc = col < N ? col : N - 1;   // clamp (stores guarded)

  const float* Ap[4];
#pragma unroll
  for (int t = 0; t < 4; ++t) {
    int row = m0 + t * 16 + idx16;
    if (row > M - 1) row = M - 1;           // clamp (stores guarded)
    Ap[t] = A + (size_t)row * Kd + khalf;
  }
  const float* Bp = B + (size_t)khalf * N + colc;

  v8f acc[4] = {};

#pragma unroll 4
  for (int k0 = 0; k0 < Kd; k0 += 4) {
    // B fragment: 4x16 f32 (lanes 0-15: rows k0,k0+1; lanes 16-31: k0+2,k0+3)
    v2f bf;
    bf.x = Bp[(size_t)k0 * N];
    bf.y = Bp[(size_t)(k0 + 1) * N];
#pragma unroll
    for (int t = 0; t < 4; ++t) {
      // A fragment: 16x4 f32, two consecutive K values per lane
      v2f af;
      af.x = Ap[t][k0];
      af.y = Ap[t][k0 + 1];
      acc[t] = __builtin_amdgcn_wmma_f32_16x16x4_f32(
          /*neg_a=*/false, af, /*neg_b=*/false, bf,
          /*c_mod=*/(short)0, acc[t], /*reuse_a=*/false, /*reuse_b=*/false);
    }
  }

  // D layout: lane<16 -> M = base + r ; lane>=16 -> M = base + r + 8
  const int mofs = (lane >> 4) * 8;
  if (col < N) {
#pragma unroll
    for (int t = 0; t < 4; ++t) {
#pragma unroll
      for (int r = 0; r < 8; ++r) {
        int m = m0 + t * 16 + mofs + r;
        if (m < M) C[(size_t)m * N + col] = acc[t][r];
      }
    }
  }
}

// ---- tvconv (9-tap per-pixel weighted patch sum) + exact-GELU gating -----
// C1: (1360, NTOT), Wgt: (12240, 196) with row = c*9+k, Hact: (680, NTOT)
__global__ __launch_bounds__(256) void k_tvconv_gate(const float* __restrict__ C1,
                                                     const float* __restrict__ Wgt,
                                                     float* __restrict__ Hact) {
  int gid = blockIdx.x * blockDim.x + threadIdx.x;
  if (gid >= HID_C * NTOT) return;
  int c = gid / NTOT;
  int n = gid % NTOT;
  int b = n / PIX, p = n % PIX;
  int h = p / 14, wc = p % 14;
  float u1 = 0.f, u2 = 0.f;
  const float* w1 = Wgt + (size_t)(c * 9) * PIX + p;
  const float* w2 = Wgt + (size_t)((c + HID_C) * 9) * PIX + p;
  const float* c1 = C1 + (size_t)c * NTOT + b * PIX;
  const float* c2 = C1 + (size_t)(c + HID_C) * NTOT + b * PIX;
#pragma unroll
  for (int k = 0; k < 9; ++k) {
    int i = k / 3, j = k % 3;
    int hh = h + i - 1, ww = wc + j - 1;
    if (hh >= 0 && hh < 14 && ww >= 0 && ww < 14) {
      int pp = hh * 14 + ww;
      u1 += w1[(size_t)k * PIX] * c1[pp];
      u2 += w2[(size_t)k * PIX] * c2[pp];
    }
  }
  float ge = 0.5f * u1 * (1.f + erff(u1 * 0.70710678118654752f));
  Hact[(size_t)c * NTOT + n] = ge * u2;
}

// ---------------------------- launcher ------------------------------------
extern "C" void kernel_launch(void* const* d_in, const int* in_sizes, int n_in,
                              void* d_out, int out_size, void* d_ws, size_t ws_size,
                              hipStream_t stream) {
  const float* x     = (const float*)d_in[0];
  const float* W_in  = (const float*)d_in[1];
  const float* posi  = (const float*)d_in[2];
  const float* w0    = (const float*)d_in[3];
  const float* g0    = (const float*)d_in[4];
  const float* b0    = (const float*)d_in[5];
  const float* w1    = (const float*)d_in[6];
  const float* g1    = (const float*)d_in[7];
  const float* b1    = (const float*)d_in[8];
  const float* w2    = (const float*)d_in[9];
  const float* g2    = (const float*)d_in[10];
  const float* b2    = (const float*)d_in[11];
  const float* wf    = (const float*)d_in[12];
  const float* W_out = (const float*)d_in[13];
  float* out = (float*)d_out;

  // workspace carve-up (floats)
  float* ws   = (float*)d_ws;
  float* Xp   = ws;                                 // 256  * 12544
  float* C1   = Xp   + (size_t)DIM_C * NTOT;        // 1360 * 12544
  float* Hact = C1   + (size_t)CH_C  * NTOT;        // 680  * 12544
  float* Out2 = Hact + (size_t)HID_C * NTOT;        // 256  * 12544
  float* t0   = Out2 + (size_t)DIM_C * NTOT;        // 64*196
  float* t1   = t0   + INTER * PIX;
  float* t2   = t1   + INTER * PIX;
  float* ttmp = t2   + INTER * PIX;
  float* Bf   = ttmp + INTER * PIX;                 // 576 * 196
  float* Wgt  = Bf   + (size_t)KIM * PIX;           // 12240 * 196

  dim3 blk(256);

  // 1) pack x for GEMM1
  k_pack_x<<<(BATCH * DIM_C * PIX + 255) / 256, blk, 0, stream>>>(x, Xp);

  // 2) weight-generating convnet (tiny, LN needs global reduce -> 1 WG each)
  k_conv_ln<4>    <<<1, blk, 0, stream>>>(posi, w0, g0, b0, ttmp, t0);
  k_conv_ln<INTER><<<1, blk, 0, stream>>>(t0,   w1, g1, b1, ttmp, t1);
  k_conv_ln<INTER><<<1, blk, 0, stream>>>(t1,   w2, g2, b2, ttmp, t2);

  // 3) im2col for final conv, then WMMA GEMMs
  k_im2col<<<(KIM * PIX + 255) / 256, blk, 0, stream>>>(t2, Bf);

  // final conv as GEMM: wf(12240x576) @ Bf(576x196) -> Wgt(12240x196)
  {
    dim3 grid((PIX + 31) / 32, (OUTF + 255) / 256);
    k_gemm_wmma<OUTF, PIX, KIM><<<grid, blk, 0, stream>>>(wf, Bf, Wgt);
  }
  // GEMM1: W_in(1360x256) @ Xp(256x12544) -> C1(1360x12544)
  {
    dim3 grid((NTOT + 31) / 32, (CH_C + 255) / 256);
    k_gemm_wmma<CH_C, NTOT, DIM_C><<<grid, blk, 0, stream>>>(W_in, Xp, C1);
  }

  // 4) tvconv + GELU gating -> Hact(680x12544)
  k_tvconv_gate<<<(HID_C * NTOT + 255) / 256, blk, 0, stream>>>(C1, Wgt, Hact);

  // 5) GEMM2: W_out(256x680) @ Hact(680x12544) -> Out2(256x12544)
  {
    dim3 grid((NTOT + 31) / 32, (DIM_C + 255) / 256);
    k_gemm_wmma<DIM_C, NTOT, HID_C><<<grid, blk, 0, stream>>>(W_out, Hact, Out2);
  }

  // 6) unpack to (b,o,h,w)
  k_unpack<<<(BATCH * DIM_C * PIX + 255) / 256, blk, 0, stream>>>(Out2, out);
}